// AffineTransformer_300647710785
// MI455X (gfx1250) — compile-verified
//
#include <hip/hip_runtime.h>

// Fixed spatial dims from the reference problem (B, C derived at launch).
constexpr int Dd = 160, Hh = 192, Ww = 224;
constexpr int HW  = Hh * Ww;
constexpr int DHW = Dd * HW;

// CT > 0: compile-time channel count (fully unrolled, gathers batched into one
//         VMEM clause). CT == 0: runtime channel loop fallback.
template <int CT>
__global__ __launch_bounds__(256)
void affine_sample3d_kernel(const float* __restrict__ src,
                            const float* __restrict__ mat,
                            float* __restrict__ out,
                            int B, int Crt)
{
    const int C  = (CT > 0) ? CT : Crt;
    const int w  = blockIdx.x * 32 + (int)threadIdx.x;   // lane -> w : coalesced
    const int h  = blockIdx.y * 8  + (int)threadIdx.y;
    const int bz = blockIdx.z;
    const int b  = bz / Dd;
    const int d  = bz - b * Dd;
    if (w >= Ww || h >= Hh) return;

    // theta = mat / norm : translation column of rows (x,y,z) divided by (D,H,W)
    // (matches the reference's norm matrix exactly).
    const float* m = mat + b * 12;                        // uniform -> s_load nv
    const float t00 = m[0], t01 = m[1], t02 = m[2],  t03 = m[3]  * (1.0f / Dd);
    const float t10 = m[4], t11 = m[5], t12 = m[6],  t13 = m[7]  * (1.0f / Hh);
    const float t20 = m[8], t21 = m[9], t22 = m[10], t23 = m[11] * (1.0f / Ww);

    // base coords in [-1,1], align_corners=True
    const float xw = fmaf((float)w, 2.0f / (Ww - 1), -1.0f);
    const float yh = fmaf((float)h, 2.0f / (Hh - 1), -1.0f);
    const float zd = fmaf((float)d, 2.0f / (Dd - 1), -1.0f);

    const float gx = fmaf(t00, xw, fmaf(t01, yh, fmaf(t02, zd, t03)));
    const float gy = fmaf(t10, xw, fmaf(t11, yh, fmaf(t12, zd, t13)));
    const float gz = fmaf(t20, xw, fmaf(t21, yh, fmaf(t22, zd, t23)));

    // unnormalize: -1 -> 0, +1 -> dim-1
    const float ix = fmaf(gx, (Ww - 1) * 0.5f, (Ww - 1) * 0.5f);
    const float iy = fmaf(gy, (Hh - 1) * 0.5f, (Hh - 1) * 0.5f);
    const float iz = fmaf(gz, (Dd - 1) * 0.5f, (Dd - 1) * 0.5f);

    const float fx0 = floorf(ix), fy0 = floorf(iy), fz0 = floorf(iz);
    const float fx = ix - fx0, fy = iy - fy0, fz = iz - fz0;
    const int x0 = (int)fx0, y0 = (int)fy0, z0 = (int)fz0;
    const int x1 = x0 + 1, y1 = y0 + 1, z1 = z0 + 1;

    // clamped indices + validity-masked 1D weights => branchless zero padding
    const int x0c = min(max(x0, 0), Ww - 1), x1c = min(max(x1, 0), Ww - 1);
    const int y0c = min(max(y0, 0), Hh - 1), y1c = min(max(y1, 0), Hh - 1);
    const int z0c = min(max(z0, 0), Dd - 1), z1c = min(max(z1, 0), Dd - 1);

    const float wx0 = (x0 >= 0 && x0 < Ww) ? (1.0f - fx) : 0.0f;
    const float wx1 = (x1 >= 0 && x1 < Ww) ?  fx         : 0.0f;
    const float wy0 = (y0 >= 0 && y0 < Hh) ? (1.0f - fy) : 0.0f;
    const float wy1 = (y1 >= 0 && y1 < Hh) ?  fy         : 0.0f;
    const float wz0 = (z0 >= 0 && z0 < Dd) ? (1.0f - fz) : 0.0f;
    const float wz1 = (z1 >= 0 && z1 < Dd) ?  fz         : 0.0f;

    // 8 final tap weights, computed ONCE, shared by all channels
    const float wzy00 = wz0 * wy0, wzy01 = wz0 * wy1;
    const float wzy10 = wz1 * wy0, wzy11 = wz1 * wy1;
    const float W000 = wzy00 * wx0, W001 = wzy00 * wx1;
    const float W010 = wzy01 * wx0, W011 = wzy01 * wx1;
    const float W100 = wzy10 * wx0, W101 = wzy10 * wx1;
    const float W110 = wzy11 * wx0, W111 = wzy11 * wx1;

    // linear tap offsets within one channel volume
    const int r00 = (z0c * Hh + y0c) * Ww;
    const int r01 = (z0c * Hh + y1c) * Ww;
    const int r10 = (z1c * Hh + y0c) * Ww;
    const int r11 = (z1c * Hh + y1c) * Ww;
    const int i000 = r00 + x0c, i001 = r00 + x1c;
    const int i010 = r01 + x0c, i011 = r01 + x1c;
    const int i100 = r10 + x0c, i101 = r10 + x1c;
    const int i110 = r11 + x0c, i111 = r11 + x1c;

    const size_t bbase = (size_t)b * (size_t)C * (size_t)DHW;
    const float* sb = src + bbase;
    float*       ob = out + bbase;
    const int opix = d * HW + h * Ww + w;

#pragma unroll
    for (int c = 0; c < ((CT > 0) ? CT : 1); ++c) {
        // generic fallback handles C channels via inner runtime loop below
        const int clim = (CT > 0) ? 1 : Crt;
        for (int cc = 0; cc < clim; ++cc) {
            const int ch = (CT > 0) ? c : cc;
            const float* __restrict__ p = sb + (size_t)ch * DHW;
            // 8 gathers; wave-coherent addresses -> coalesced lines, L2 reuse
            const float v000 = p[i000];
            const float v001 = p[i001];
            const float v010 = p[i010];
            const float v011 = p[i011];
            const float v100 = p[i100];
            const float v101 = p[i101];
            const float v110 = p[i110];
            const float v111 = p[i111];

            float acc =      v000 * W000;
            acc = fmaf(v001, W001, acc);
            acc = fmaf(v010, W010, acc);
            acc = fmaf(v011, W011, acc);
            acc = fmaf(v100, W100, acc);
            acc = fmaf(v101, W101, acc);
            acc = fmaf(v110, W110, acc);
            acc = fmaf(v111, W111, acc);

            // Non-temporal store (TH=NT): keep L2 for the src gather stream.
            __builtin_nontemporal_store(acc, ob + (size_t)ch * DHW + opix);
        }
    }
}

extern "C" void kernel_launch(void* const* d_in, const int* in_sizes, int n_in,
                              void* d_out, int out_size, void* d_ws, size_t ws_size,
                              hipStream_t stream)
{
    (void)n_in; (void)out_size; (void)d_ws; (void)ws_size;
    const float* src = (const float*)d_in[0];
    const float* mat = (const float*)d_in[1];
    float*       out = (float*)d_out;

    const int B = in_sizes[1] / 12;                                   // mat is [B,3,4]
    const int C = (int)((long long)in_sizes[0] / ((long long)B * DHW));

    dim3 block(32, 8, 1);                               // 8 wave32s; lane == w
    dim3 grid(Ww / 32, Hh / 8, (unsigned)(B * Dd));     // 7 x 24 x 320, no tails

    if (C == 2) {       // reference shape: compile-time unrolled, batched gathers
        affine_sample3d_kernel<2><<<grid, block, 0, stream>>>(src, mat, out, B, C);
    } else if (C == 1) {
        affine_sample3d_kernel<1><<<grid, block, 0, stream>>>(src, mat, out, B, C);
    } else if (C == 4) {
        affine_sample3d_kernel<4><<<grid, block, 0, stream>>>(src, mat, out, B, C);
    } else {
        affine_sample3d_kernel<0><<<grid, block, 0, stream>>>(src, mat, out, B, C);
    }
}